// ICS_GNN_2130303779150
// MI455X (gfx1250) — compile-verified
//
#include <hip/hip_runtime.h>

typedef __attribute__((ext_vector_type(2))) float v2f;
typedef __attribute__((ext_vector_type(8))) float v8f;

#define GNN_N_NODES 100000
#define GNN_N_EDGES 3200000

// ---------------------------------------------------------------- utilities
__global__ void gnn_fill_zero(float* __restrict__ p, int n) {
    int i = blockIdx.x * blockDim.x + threadIdx.x;
    if (i < n) p[i] = 0.0f;
}

__global__ void gnn_degree_kernel(const int* __restrict__ dst,
                                  float* __restrict__ deg, int nE) {
    int e = blockIdx.x * blockDim.x + threadIdx.x;
    if (e < nE) atomicAdd(&deg[dst[e]], 1.0f);
}

// ------------------------------------------------- edge gather + scatter-add
// One thread per (edge, 4-float chunk). Gather h[src] as float4 (coalesced,
// L2-resident: node table is 25.6MB << 192MB L2), scatter with f32 atomics
// (non-returning -> global_atomic_add_f32, STOREcnt path).
template <int D>
__global__ void gnn_scatter_kernel(const float* __restrict__ h,
                                   const int* __restrict__ src,
                                   const int* __restrict__ dst,
                                   float* __restrict__ agg, int nE) {
    constexpr int CH = D / 4;
    long long tid = (long long)blockIdx.x * blockDim.x + threadIdx.x;
    long long total = (long long)nE * CH;
    if (tid >= total) return;
    int e = (int)(tid / CH);
    int c = (int)(tid % CH);
    int s = src[e];
    int d = dst[e];
    const float4 v = ((const float4*)(h + (long long)s * D))[c];
    float* out = agg + (long long)d * D + c * 4;
    atomicAdd(out + 0, v.x);
    atomicAdd(out + 1, v.y);
    atomicAdd(out + 2, v.z);
    atomicAdd(out + 3, v.w);
}

// ------------------------------------------------------ fused SAGE layer GEMM
// h_out[16-node tile] = relu( (agg/deg) @ Wl + bl + h_in @ Wr )
// One wave per (16-node tile, 16-output-col tile), V_WMMA_F32_16X16X4_F32
// accumulating over K in steps of 4 for both Wl and Wr terms.
//
// VGPR layouts per CDNA5 ISA 7.12.2:
//   A 16x4 f32 : lanes 0-15 row=lane K={0,1} in v0,v1 ; lanes 16-31 K={2,3}
//   B 4x16 f32 : lanes 0-15 col=lane K={0,1} ; lanes 16-31 col=lane-16 K={2,3}
//   C 16x16 f32: vgpr v -> row (lane<16 ? v : v+8), col = lane&15
template <int DIN, int DOUT>
__global__ void gnn_sage_wmma_kernel(const float* __restrict__ hin,
                                     const float* __restrict__ agg,
                                     const float* __restrict__ deg,
                                     const float* __restrict__ Wl,
                                     const float* __restrict__ bl,
                                     const float* __restrict__ Wr,
                                     float* __restrict__ hout, int nNodes) {
    constexpr int COLT = DOUT / 16;           // output col tiles
    constexpr int TILES_PER_BLOCK = 8 / COLT; // 8 waves / block (256 thr, wave32)
    const int wave = threadIdx.x >> 5;
    const int lane = threadIdx.x & 31;
    const int colTile = wave % COLT;
    const int nodeTile = blockIdx.x * TILES_PER_BLOCK + (wave / COLT);
    const int nodeBase = nodeTile * 16;
    if (nodeBase >= nNodes) return;          // wave-uniform: EXEC stays all-1s

    const int r16 = lane & 15;               // A row / B col / C col
    const int khalf = (lane >> 4) * 2;       // lane-half K offset {0,2}

    const int anode = nodeBase + r16;
    const float invd = 1.0f / fmaxf(deg[anode], 1.0f);
    const float* __restrict__ arow = agg + (long long)anode * DIN;
    const float* __restrict__ xrow = hin + (long long)anode * DIN;
    const int ncol = colTile * 16 + r16;

    v8f acc = {};
#pragma unroll
    for (int k = 0; k < DIN; k += 4) {
        v2f a, b;
        // mean @ Wl
        a.x = arow[k + khalf + 0] * invd;
        a.y = arow[k + khalf + 1] * invd;
        b.x = Wl[(k + khalf + 0) * DOUT + ncol];
        b.y = Wl[(k + khalf + 1) * DOUT + ncol];
        acc = __builtin_amdgcn_wmma_f32_16x16x4_f32(false, a, false, b,
                                                    (short)0, acc, false, false);
        // x @ Wr
        a.x = xrow[k + khalf + 0];
        a.y = xrow[k + khalf + 1];
        b.x = Wr[(k + khalf + 0) * DOUT + ncol];
        b.y = Wr[(k + khalf + 1) * DOUT + ncol];
        acc = __builtin_amdgcn_wmma_f32_16x16x4_f32(false, a, false, b,
                                                    (short)0, acc, false, false);
    }

    const float bias = bl[ncol];
    const int mbase = nodeBase + ((lane >> 4) * 8);
#pragma unroll
    for (int v = 0; v < 8; ++v) {
        float r = fmaxf(acc[v] + bias, 0.0f);
        hout[(long long)(mbase + v) * DOUT + ncol] = r;
    }
}

// ------------------------------------------------------------- MLP head
// out[i] = b5 + relu(h3[i] @ W4 + b4) @ W5 ; 32->16->1, trivially small.
__global__ void gnn_head_kernel(const float* __restrict__ h3,
                                const float* __restrict__ W4,
                                const float* __restrict__ b4,
                                const float* __restrict__ W5,
                                const float* __restrict__ b5,
                                float* __restrict__ out, int n) {
    int i = blockIdx.x * blockDim.x + threadIdx.x;
    if (i >= n) return;
    float xv[32];
    const float* __restrict__ x = h3 + (long long)i * 32;
#pragma unroll
    for (int k = 0; k < 32; ++k) xv[k] = x[k];
    float acc = b5[0];
#pragma unroll
    for (int j = 0; j < 16; ++j) {
        float hj = b4[j];
#pragma unroll
        for (int k = 0; k < 32; ++k) hj += xv[k] * W4[k * 16 + j];
        acc += fmaxf(hj, 0.0f) * W5[j];
    }
    out[i] = acc;
}

// ---------------------------------------------------------------- launcher
extern "C" void kernel_launch(void* const* d_in, const int* in_sizes, int n_in,
                              void* d_out, int out_size, void* d_ws, size_t ws_size,
                              hipStream_t stream) {
    const int N = GNN_N_NODES;
    const int E = GNN_N_EDGES;

    const float* x   = (const float*)d_in[0];
    const int* edge  = (const int*)d_in[1];
    const int* src   = edge;          // edge_index[0]
    const int* dst   = edge + E;      // edge_index[1]
    const float* W1l = (const float*)d_in[2];
    const float* b1  = (const float*)d_in[3];
    const float* W1r = (const float*)d_in[4];
    const float* W2l = (const float*)d_in[5];
    const float* b2  = (const float*)d_in[6];
    const float* W2r = (const float*)d_in[7];
    const float* W3l = (const float*)d_in[8];
    const float* b3  = (const float*)d_in[9];
    const float* W3r = (const float*)d_in[10];
    const float* W4  = (const float*)d_in[11];
    const float* b4  = (const float*)d_in[12];
    const float* W5  = (const float*)d_in[13];
    const float* b5  = (const float*)d_in[14];
    float* out = (float*)d_out;

    // workspace: deg[N] | agg[N*64] | bufA[N*64] | bufB[N*64]  (~77 MB)
    float* deg  = (float*)d_ws;
    float* agg  = deg + N;
    float* bufA = agg + (size_t)N * 64;
    float* bufB = bufA + (size_t)N * 64;

    const int T = 256;
    auto blocks = [](long long n, int t) { return (int)((n + t - 1) / t); };

    // degrees (layer-invariant, recomputed each call for determinism)
    gnn_fill_zero<<<blocks(N, T), T, 0, stream>>>(deg, N);
    gnn_degree_kernel<<<blocks(E, T), T, 0, stream>>>(dst, deg, E);

    // ---- layer 1: 4 -> 64
    gnn_fill_zero<<<blocks((long long)N * 4, T), T, 0, stream>>>(agg, N * 4);
    gnn_scatter_kernel<4><<<blocks((long long)E * 1, T), T, 0, stream>>>(x, src, dst, agg, E);
    gnn_sage_wmma_kernel<4, 64><<<N / 32, T, 0, stream>>>(x, agg, deg, W1l, b1, W1r, bufA, N);

    // ---- layer 2: 64 -> 64
    gnn_fill_zero<<<blocks((long long)N * 64, T), T, 0, stream>>>(agg, N * 64);
    gnn_scatter_kernel<64><<<blocks((long long)E * 16, T), T, 0, stream>>>(bufA, src, dst, agg, E);
    gnn_sage_wmma_kernel<64, 64><<<N / 32, T, 0, stream>>>(bufA, agg, deg, W2l, b2, W2r, bufB, N);

    // ---- layer 3: 64 -> 32
    gnn_fill_zero<<<blocks((long long)N * 64, T), T, 0, stream>>>(agg, N * 64);
    gnn_scatter_kernel<64><<<blocks((long long)E * 16, T), T, 0, stream>>>(bufB, src, dst, agg, E);
    gnn_sage_wmma_kernel<64, 32><<<blocks(N, 64), T, 0, stream>>>(bufB, agg, deg, W3l, b3, W3r, bufA, N);

    // ---- head: 32 -> 16 -> 1
    gnn_head_kernel<<<blocks(N, T), T, 0, stream>>>(bufA, W4, b4, W5, b5, out, N);
}